// CogRNN_764504179399
// MI455X (gfx1250) — compile-verified
//
#include <hip/hip_runtime.h>
#include <hip/hip_bf16.h>
#include <math.h>

// ---- problem constants (match reference) ----
#define T_LEN   2048
#define B_DIM   8
#define F_DIM   64
#define BF      (B_DIM * F_DIM)        // 512
#define KK      8
#define NN      66                      // N_TAUS + 2K
#define NTAUS   50
#define NPAD    68                      // K-dim padded to multiple of 4
#define MPAD    64                      // m-tiles: 4 x 16 (50 valid)
#define DT_F    0.05f
#define TSTR_MIN 0.1
#define TSTR_MAX 100.0

typedef __attribute__((ext_vector_type(2))) float v2f;
typedef __attribute__((ext_vector_type(8))) float v8f;

// ======================================================================
// Init kernel: build -s[n] (66 floats) and POST slice padded [68][64]
// into workspace. One block, doubles throughout.
//   POST[n,m] = (D^8)[m,n] * tau[m] * s[m]^9 / 8!      (K=8, G=1)
// row m of D^8 == (D^T)^8 applied to e_m.  D is tridiagonal.
// ======================================================================
__global__ void init_consts_kernel(float* __restrict__ s_neg,
                                   float* __restrict__ postpad)
{
    const int tid = threadIdx.x;
    const double c    = pow(TSTR_MAX / TSTR_MIN, 1.0 / (NTAUS - 1));
    const double invc = 1.0 / c;

    // zero the padded POST tile
    for (int i = tid; i < NPAD * MPAD; i += blockDim.x) postpad[i] = 0.0f;
    __syncthreads();

    // s values (shared so every thread can read all of them cheaply)
    __shared__ double s_sh[NN];
    if (tid < NN) {
        double tau = TSTR_MIN * pow(c, (double)(tid - KK));
        double s   = (double)KK / tau;
        s_sh[tid]  = s;
        s_neg[tid] = -(float)s;          // DT_SCALE == 1
    }
    __syncthreads();

    // threads 0..49 each build POST column for m = tid + 8
    if (tid < NTAUS) {
        const int m = tid + KK;
        double v[NN], w[NN];
        for (int i = 0; i < NN; ++i) v[i] = 0.0;
        v[m] = 1.0;
        for (int it = 0; it < KK; ++it) {
            for (int i = 0; i < NN; ++i) {
                double acc = 0.0;
                // w = D^T v ; row r of D valid for 1 <= r <= NN-2
                if (i - 1 >= 1 && i - 1 <= NN - 2) {       // D[i-1][i] = hi(i-1)
                    double denom = s_sh[i] - s_sh[i - 2];
                    acc += (c / denom) * v[i - 1];
                }
                if (i >= 1 && i <= NN - 2) {               // D[i][i] = diag(i)
                    double denom = s_sh[i + 1] - s_sh[i - 1];
                    acc += ((invc - c) / denom) * v[i];
                }
                if (i + 1 <= NN - 2) {                     // D[i+1][i] = lo(i+1)
                    double denom = s_sh[i + 2] - s_sh[i];
                    acc += (-invc / denom) * v[i + 1];
                }
                w[i] = acc;
            }
            for (int i = 0; i < NN; ++i) v[i] = w[i];
        }
        // scale: tau[m]^G * exp(-lgamma(9) + 9*log(s[m])) = tau*s^9/40320
        double tau_m  = (double)KK / s_sh[m];
        double scale  = tau_m * pow(s_sh[m], (double)(KK + 1)) / 40320.0;
        for (int n = 0; n < NN; ++n)
            postpad[n * MPAD + tid] = (float)(v[n] * scale);
    }
}

// ======================================================================
// Main kernel: one block per (b,f); 128 threads = 4 waves.
// Lanes 0..65 run the exp-space recurrence; every 16 steps the window
// is staged in LDS and each wave computes one 16-wide m-tile of til_f
// with 17 chained V_WMMA_F32_16X16X4_F32. All global output traffic
// (til_f and F_out) is drained in the all-waves phase after the
// barrier so the serial scan loop stays free of exec juggling.
// ======================================================================
#define TILE_STRIDE 69   // odd-ish stride -> 16 rows hit distinct LDS banks

__global__ void __launch_bounds__(128)
scan_wmma_kernel(const float* __restrict__ f_in,
                 const float* __restrict__ alpha,
                 const float* __restrict__ delta,
                 const float* __restrict__ s_neg,
                 const float* __restrict__ postpad,
                 float* __restrict__ til_f,
                 float* __restrict__ h_out,
                 float* __restrict__ F_out)
{
    const int bf   = blockIdx.x;          // b*F + f
    const int tid  = threadIdx.x;
    const int lane = tid & 31;
    const int wave = tid >> 5;

    __shared__ float tile[16][TILE_STRIDE];   // F_exp window [t_local][n]
    __shared__ float sc[3][16];               // staged alpha / delta / f

    // --- preload B operands (POST columns for this wave's m-tile) ---
    // Assumed 32-bit 4x16 B layout (mirror of the documented 16x4 A):
    //   V0: lanes0-15 = row k0+0, lanes16-31 = row k0+2
    //   V1: lanes0-15 = row k0+1, lanes16-31 = row k0+3
    v2f Bop[17];
    {
        const int col   = wave * 16 + (lane & 15);
        const int khalf = (lane < 16) ? 0 : 2;
#pragma unroll
        for (int kc = 0; kc < 17; ++kc) {
            const int k0 = kc * 4 + khalf;
            Bop[kc].x = postpad[(k0 + 0) * MPAD + col];
            Bop[kc].y = postpad[(k0 + 1) * MPAD + col];
        }
    }

    // zero the K-pad columns (66,67) ONCE; nothing overwrites them later
    if (tid >= NN && tid < NPAD) {
#pragma unroll
        for (int i = 0; i < 16; ++i) tile[i][tid] = 0.0f;
    }

    const bool scan_lane = (tid < NN);
    const float neg_s = scan_lane ? s_neg[tid] : 0.0f;
    float Fst = 0.0f;                         // exp(-inf) = 0

    for (int t0 = 0; t0 < T_LEN; t0 += 16) {
        // stage the 16 per-(b,f) scalars for this window
        if (tid < 48) {
            const int which = tid >> 4, i = tid & 15;
            const float* src = (which == 0) ? alpha : (which == 1) ? delta : f_in;
            sc[which][i] = src[(size_t)(t0 + i) * BF + bf];
        }
        __syncthreads();

        // --- serial scan: pure fma/exp/ds_store, no global traffic ---
        if (scan_lane) {
            const int n = tid;
#pragma unroll
            for (int i = 0; i < 16; ++i) {
                const float a = sc[0][i], d = sc[1][i], fv = sc[2][i];
                const float ll = fmaf(a, DT_F, d) * neg_s;   // log decay
                Fst = fmaf(Fst, __expf(ll), fv * DT_F);      // exp-space scan
                tile[i][n] = Fst;
            }
        }
        __syncthreads();

        // --- WMMA: D(16t x 16m) += A(16t x 4n) * B(4n x 16m), 17 chunks ---
        v8f acc = {0.f, 0.f, 0.f, 0.f, 0.f, 0.f, 0.f, 0.f};
        const int row   = lane & 15;
        const int khalf = (lane < 16) ? 0 : 2;
#pragma unroll
        for (int kc = 0; kc < 17; ++kc) {
            v2f a;
            a.x = tile[row][kc * 4 + khalf + 0];
            a.y = tile[row][kc * 4 + khalf + 1];
            acc = __builtin_amdgcn_wmma_f32_16x16x4_f32(
                false, a, false, Bop[kc], (short)0, acc, false, false);
        }

        // store this wave's 16x16 tile of til_f (C/D layout: VGPR r holds
        // row M=r for lanes 0-15, row M=r+8 for lanes 16-31)
        {
            const int m = wave * 16 + (lane & 15);
            if (m < NTAUS) {
#pragma unroll
                for (int r = 0; r < 8; ++r) {
                    const int tl = (lane < 16) ? r : r + 8;
                    til_f[((size_t)(t0 + tl) * BF + bf) * NTAUS + m] = acc[r];
                }
            }
        }

        // cooperative drain of F_out (16 x 50 block) from the LDS tile
        for (int idx = tid; idx < 16 * NTAUS; idx += 128) {
            const int i  = idx / NTAUS;
            const int mc = idx - i * NTAUS;
            F_out[((size_t)(t0 + i) * BF + bf) * NTAUS + mc] = tile[i][mc + KK];
        }
        // no extra barrier needed: next window's tile writes happen only
        // after the next __syncthreads(), which orders them after all
        // tile reads (WMMA + F_out drain) of this window.
    }

    if (scan_lane)
        h_out[(size_t)bf * NN + tid] = __logf(Fst);
}

// ======================================================================
extern "C" void kernel_launch(void* const* d_in, const int* in_sizes, int n_in,
                              void* d_out, int out_size, void* d_ws, size_t ws_size,
                              hipStream_t stream)
{
    const float* f_in  = (const float*)d_in[0];
    const float* alpha = (const float*)d_in[1];
    const float* delta = (const float*)d_in[2];

    float* out   = (float*)d_out;
    float* til_f = out;                                   // (T,B,F,50)
    float* h_out = out + (size_t)T_LEN * BF * NTAUS;      // (B,F,66)
    float* F_out = h_out + (size_t)BF * NN;               // (T,B,F,50)

    float* s_neg   = (float*)d_ws;                        // 66 floats
    float* postpad = (float*)d_ws + 128;                  // [68][64] floats

    init_consts_kernel<<<1, 128, 0, stream>>>(s_neg, postpad);
    scan_wmma_kernel<<<BF, 128, 0, stream>>>(f_in, alpha, delta,
                                             s_neg, postpad,
                                             til_f, h_out, F_out);
}